// JointLoss_49563922596543
// MI455X (gfx1250) — compile-verified
//
#include <hip/hip_runtime.h>

// ---------------------------------------------------------------------------
// JointLoss for MI455X (gfx1250, wave32).
// Bandwidth-bound streaming reduction (~129 MB @ 23.3 TB/s => ~5.5 us floor).
// All block-level reductions are done with V_WMMA_F32_16X16X4_F32 (exact f32
// matrix op): D = A x ones + C accumulates row sums, 64 f32 adds per WMMA.
// ---------------------------------------------------------------------------

typedef __attribute__((ext_vector_type(2))) float v2f;
typedef __attribute__((ext_vector_type(8))) float v8f;

#define B_      16
#define HW_     (512 * 512)
#define NEQ_    5000
#define NINEQ_  5000
#define TAU_    0.425f
#define NBLK2   1024

// Deterministic 256-thread block reduction via the matrix pipe.
// lds must have room for 257 floats. Returns the total to ALL threads.
__device__ __forceinline__ float block_reduce_wmma_256(float v, float* lds) {
    const int tid = threadIdx.x;
    lds[tid] = v;
    __syncthreads();
    if (tid < 32) {  // wave 0 fully active -> EXEC == all ones (wave32)
        v8f acc = {0.f, 0.f, 0.f, 0.f, 0.f, 0.f, 0.f, 0.f};
        v2f ones = {1.0f, 1.0f};  // B = 4x16 all-ones => D[m][n] = rowsum_m + C
        #pragma unroll
        for (int i = 0; i < 4; ++i) {
            v2f a = {lds[i * 64 + 2 * tid], lds[i * 64 + 2 * tid + 1]};
            // v_wmma_f32_16x16x4_f32 : exact f32 MAC, chained on C.
            acc = __builtin_amdgcn_wmma_f32_16x16x4_f32(
                false, a, false, ones, (short)0, acc, false, false);
        }
        // Every column of D holds identical row sums. Lanes 0-15 see rows
        // M=r (VGPR r), lanes 16-31 see rows M=8+r -> xor-16 pairs cover all M.
        float s = acc[0] + acc[1] + acc[2] + acc[3] +
                  acc[4] + acc[5] + acc[6] + acc[7];
        s += __shfl_xor(s, 16, 32);
        if (tid == 0) lds[256] = s;
    }
    __syncthreads();
    float r = lds[256];
    __syncthreads();
    return r;
}

__device__ __forceinline__ int lin_idx_(float y, float x) {
    int yi = (int)floorf(y * 512.0f);
    int xi = (int)floorf(x * 512.0f);
    return yi * 512 + xi;
}

// One block per batch image. Gathers hit L2 (whole R = 16 MB << 192 MB L2).
__global__ __launch_bounds__(256)
void ranking_kernel(const float* __restrict__ R,
                    const float* __restrict__ je,
                    const float* __restrict__ ji,
                    float* __restrict__ ws_rank) {
    __shared__ float lds[257];
    const int b = blockIdx.x;
    const float* Rb  = R  + (size_t)b * HW_;
    const float* jeb = je + (size_t)b * NEQ_ * 5;
    const float* jib = ji + (size_t)b * NINEQ_ * 5;

    float eq = 0.0f, iq = 0.0f, nv = 0.0f;
    for (int j = threadIdx.x; j < NEQ_; j += blockDim.x) {
        const float* rec = jeb + j * 5;
        float p1 = Rb[lin_idx_(rec[0], rec[1])];
        float p2 = Rb[lin_idx_(rec[2], rec[3])];
        float d = p1 - p2;
        eq += rec[4] * d * d;
    }
    for (int j = threadIdx.x; j < NINEQ_; j += blockDim.x) {
        const float* rec = jib + j * 5;
        float q1 = Rb[lin_idx_(rec[0], rec[1])];
        float q2 = Rb[lin_idx_(rec[2], rec[3])];
        float d = q2 - q1 + TAU_;           // relu arg; d>=0 <=> valid pair
        if (d >= 0.0f) { nv += 1.0f; iq += rec[4] * d * d; }
    }
    float eqs = block_reduce_wmma_256(eq, lds);
    float iqs = block_reduce_wmma_256(iq, lds);
    float nvs = block_reduce_wmma_256(nv, lds);
    if (threadIdx.x == 0) {
        ws_rank[b] = eqs / (5000.0f + 1e-8f) + iqs / (nvs + 1e-8f);
    }
}

// Fused streaming pass: recon = sum((rgb - chrom*R*S)^2), sp = sum((S-0.5)^2).
// float4 (b128) loads from all 8 planes; grid-stride, 4 groups per thread.
__global__ __launch_bounds__(256)
void recon_sp_kernel(const float* __restrict__ R,
                     const float* __restrict__ S,
                     const float* __restrict__ rgb,
                     const float* __restrict__ chrom,
                     float* __restrict__ ws_recon,
                     float* __restrict__ ws_sp) {
    __shared__ float lds[257];
    const int tid = blockIdx.x * blockDim.x + threadIdx.x;
    const int nth = gridDim.x * blockDim.x;
    const int G = (B_ * HW_) / 4;  // float4 groups over (B,1,H,W)

    const float4* R4  = (const float4*)R;
    const float4* S4  = (const float4*)S;
    const float4* T4  = (const float4*)rgb;
    const float4* M4  = (const float4*)chrom;

    float recon = 0.0f, sp = 0.0f;
    for (int g = tid; g < G; g += nth) {
        const int b   = g >> 16;        // HW/4 = 65536 float4 groups per plane
        const int hw4 = g & 0xFFFF;
        const float4 r = R4[g];
        const float4 s = S4[g];
        const float4 rs = make_float4(r.x * s.x, r.y * s.y, r.z * s.z, r.w * s.w);
        const int base = ((b * 3) << 16) + hw4;
        #pragma unroll
        for (int c = 0; c < 3; ++c) {
            const float4 t = T4[base + (c << 16)];
            const float4 m = M4[base + (c << 16)];
            const float dx = t.x - m.x * rs.x;
            const float dy = t.y - m.y * rs.y;
            const float dz = t.z - m.z * rs.z;
            const float dw = t.w - m.w * rs.w;
            recon += dx * dx + dy * dy + dz * dz + dw * dw;
        }
        const float ex = s.x - 0.5f, ey = s.y - 0.5f;
        const float ez = s.z - 0.5f, ew = s.w - 0.5f;
        sp += ex * ex + ey * ey + ez * ez + ew * ew;
    }
    const float rt = block_reduce_wmma_256(recon, lds);
    const float st = block_reduce_wmma_256(sp, lds);
    if (threadIdx.x == 0) {
        ws_recon[blockIdx.x] = rt;
        ws_sp[blockIdx.x]    = st;
    }
}

__global__ __launch_bounds__(256)
void finalize_kernel(const float* __restrict__ ws_rank,
                     const float* __restrict__ ws_recon,
                     const float* __restrict__ ws_sp,
                     float* __restrict__ out) {
    __shared__ float lds[257];
    float recon = 0.0f, sp = 0.0f;
    for (int i = threadIdx.x; i < NBLK2; i += 256) {
        recon += ws_recon[i];
        sp    += ws_sp[i];
    }
    const float rt = block_reduce_wmma_256(recon, lds);
    const float st = block_reduce_wmma_256(sp, lds);
    if (threadIdx.x == 0) {
        float rank = 0.0f;
        for (int i = 0; i < B_; ++i) rank += ws_rank[i];
        rank *= (1.0f / (float)B_);
        const float reconm = rt / (float)(B_ * 3 * HW_);  // 12582912, exact f32
        const float spm    = st / (float)(B_ * HW_);      // 4194304,  exact f32
        out[0] = 4.0f * rank + 2.0f * reconm + 0.25f * spm;
    }
}

extern "C" void kernel_launch(void* const* d_in, const int* in_sizes, int n_in,
                              void* d_out, int out_size, void* d_ws, size_t ws_size,
                              hipStream_t stream) {
    (void)in_sizes; (void)n_in; (void)out_size; (void)ws_size;
    const float* R     = (const float*)d_in[0];
    const float* S     = (const float*)d_in[1];
    const float* rgb   = (const float*)d_in[2];
    const float* chrom = (const float*)d_in[3];
    const float* je    = (const float*)d_in[4];
    const float* ji    = (const float*)d_in[5];
    float* out = (float*)d_out;

    float* ws       = (float*)d_ws;
    float* ws_rank  = ws;                 // 16 floats
    float* ws_recon = ws + 16;            // 1024 floats
    float* ws_sp    = ws + 16 + NBLK2;    // 1024 floats

    hipLaunchKernelGGL(ranking_kernel,  dim3(B_),    dim3(256), 0, stream,
                       R, je, ji, ws_rank);
    hipLaunchKernelGGL(recon_sp_kernel, dim3(NBLK2), dim3(256), 0, stream,
                       R, S, rgb, chrom, ws_recon, ws_sp);
    hipLaunchKernelGGL(finalize_kernel, dim3(1),     dim3(256), 0, stream,
                       ws_rank, ws_recon, ws_sp, out);
}